// TemporalEvolver_85220741087616
// MI455X (gfx1250) — compile-verified
//
#include <hip/hip_runtime.h>
#include <cstdint>
#include <cstddef>

// TemporalEvolver: psi_{t+1} = (1 + LAMB - 2*KAPPA)*psi + KAPPA*(psi[n-1]+psi[n+1])
// on a circular ring of 2^20 nodes; emit out[n][t] for t=0..127 (fp32, 512MB).
// Pure store-bandwidth problem (~22us at 23.3 TB/s). LDS time-marching scan with
// halo = 127, transpose-staged non-temporal streaming stores, async-to-LDS input.

namespace {
constexpr int   kNodes  = 1 << 20;          // NUM_NODES
constexpr int   kT      = 128;              // NUM_T
constexpr int   kHalo   = kT - 1;           // 127
constexpr int   kTile   = 256;              // nodes per block
constexpr int   kWin    = kTile + 2*kHalo;  // 510 values loaded per block
constexpr int   kBuf    = 512;
constexpr int   kTch    = 32;               // time columns staged per flush
constexpr int   kStride = kTch + 1;         // 33: gcd(33,64)=1 -> conflict-free columns
constexpr float kKappa  = 0.1f;
constexpr float kLamb   = 0.05f;
}

typedef float v4f __attribute__((ext_vector_type(4)));
typedef __attribute__((address_space(1))) int  g_i32;
typedef __attribute__((address_space(3))) int  l_i32;

#if __has_builtin(__builtin_amdgcn_global_load_async_to_lds_b32)
#define ASYNC_LDS 1
#else
#define ASYNC_LDS 0
#endif

__global__ __launch_bounds__(kTile) void evolve_kernel(
    const float* __restrict__ field, float* __restrict__ out) {
  __shared__ float psi[2][kBuf];            // ping-pong state window
  __shared__ float outs[kTile][kStride];    // transpose staging, padded

  const int tid  = threadIdx.x;
  const int base = blockIdx.x * kTile;

  // ---- Load psi0 window: nodes [base-127, base+256+127), circular wrap. ----
  // field[g][0] lives at byte offset g*512 (only column 0 of the input is used).
#pragma unroll
  for (int h = 0; h < 2; ++h) {
    const int k = tid + h * kTile;
    if (k < kWin) {
      const int g = (base + k - kHalo) & (kNodes - 1);
      const float* src = field + (size_t)g * kT;
#if ASYNC_LDS
      __builtin_amdgcn_global_load_async_to_lds_b32(
          (g_i32*)(uintptr_t)src,
          (l_i32*)&psi[0][k],
          0, 0);
#else
      psi[0][k] = *src;
#endif
    }
  }
#if ASYNC_LDS
#if __has_builtin(__builtin_amdgcn_s_wait_asynccnt)
  __builtin_amdgcn_s_wait_asynccnt(0);
#else
  asm volatile("s_wait_asynccnt 0" ::: "memory");
#endif
#endif
  __syncthreads();

  // t = 0 column is psi0 itself.
  outs[tid][0] = psi[0][kHalo + tid];

  int cur = 0;
  for (int t = 1; t < kT; ++t) {
    const float* __restrict__ p  = psi[cur];
    float*       __restrict__ nx = psi[cur ^ 1];
    // Update interior [1, 509). Valid data window shrinks 1/side/step, so
    // boundary garbage never propagates into the tile region [127, 383).
    {
      const int j = 1 + tid;                        // 1..256
      const float c = p[j];
      nx[j] = c + kKappa * ((p[j-1] + p[j+1]) - 2.0f * c) + kLamb * c;
    }
    if (tid < kWin - kTile - 2) {                   // tid < 252 -> j = 257..508
      const int j = 1 + kTile + tid;
      const float c = p[j];
      nx[j] = c + kKappa * ((p[j-1] + p[j+1]) - 2.0f * c) + kLamb * c;
    }
    __syncthreads();
    outs[tid][t & (kTch - 1)] = nx[kHalo + tid];

    if ((t & (kTch - 1)) == kTch - 1) {
      __syncthreads();
      // Flush the staged [256 nodes x 32 times] slab, node-major & coalesced:
      // 8 consecutive lanes cover one node row = 128B contiguous per row.
      const int tbase = (t >> 5) << 5;              // chunk * 32
#pragma unroll
      for (int kk = 0; kk < 8; ++kk) {
        const int q  = tid + kk * kTile;            // float4 index 0..2047
        const int r  = q >> 3;                      // node row 0..255
        const int c4 = (q & 7) << 2;                // col within chunk
        v4f v;
        v.x = outs[r][c4 + 0];
        v.y = outs[r][c4 + 1];
        v.z = outs[r][c4 + 2];
        v.w = outs[r][c4 + 3];
        v4f* dst = (v4f*)(out + (size_t)(base + r) * kT + tbase + c4);
        __builtin_nontemporal_store(v, dst);        // stream, bypass L2 retention
      }
      __syncthreads();
    }
    cur ^= 1;
  }
}

extern "C" void kernel_launch(void* const* d_in, const int* in_sizes, int n_in,
                              void* d_out, int out_size, void* d_ws, size_t ws_size,
                              hipStream_t stream) {
  (void)in_sizes; (void)n_in; (void)d_ws; (void)ws_size; (void)out_size;
  const float* field = (const float*)d_in[0];
  float* out = (float*)d_out;
  hipLaunchKernelGGL(evolve_kernel, dim3(kNodes / kTile), dim3(kTile), 0, stream,
                     field, out);
}